// KnowledgeFussion_17892833755750
// MI455X (gfx1250) — compile-verified
//
#include <hip/hip_runtime.h>
#include <hip/hip_bf16.h>
#include <math.h>

#define BS   64
#define SEQ  512
#define EMB  1024
#define NH   16
#define HD   64
#define QSCALE 0.125f   // 1/sqrt(64)

typedef __attribute__((ext_vector_type(2))) float v2f;
typedef __attribute__((ext_vector_type(8))) float v8f;

// D(16x16,f32) += A(16x4,f32) * B(4x16,f32)
__device__ __forceinline__ v8f wmma_f32(v2f a, v2f b, v8f c) {
    return __builtin_amdgcn_wmma_f32_16x16x4_f32(
        /*neg_a=*/false, a, /*neg_b=*/false, b,
        /*c_mod=*/(short)0, c, /*reuse_a=*/false, /*reuse_b=*/false);
}

// ---------------------------------------------------------------- K1: prep
// mask_pos[b] = first s with ids==MASK (0 if none); q_in[b,:] = kv_src[b,mask_pos]
__global__ void k_prep(const int* __restrict__ ids, const float* __restrict__ model,
                       const float* __restrict__ know, const int* __restrict__ mask_tok,
                       const int* __restrict__ know_tok, int* __restrict__ mask_pos,
                       float* __restrict__ q_in) {
    int b = blockIdx.x;
    int mtok = mask_tok[0], ktok = know_tok[0];
    __shared__ int sp;
    if (threadIdx.x == 0) sp = SEQ;
    __syncthreads();
    for (int s = threadIdx.x; s < SEQ; s += blockDim.x)
        if (ids[b * SEQ + s] == mtok) atomicMin(&sp, s);
    __syncthreads();
    int p = (sp < SEQ) ? sp : 0;
    if (threadIdx.x == 0) mask_pos[b] = p;
    const float* src = (ids[b * SEQ + p] == ktok)
                     ? (know + (size_t)b * EMB)
                     : (model + ((size_t)b * SEQ + p) * EMB);
    for (int e = threadIdx.x; e < EMB; e += blockDim.x)
        q_in[(size_t)b * EMB + e] = src[e];
}

// ------------------------------------------------- K2: Out = A @ W^T + bias
// A: M x K row-major, W: N x K row-major. grid = (M/16, N/16), block = 32.
__global__ void k_gemm_abt(const float* __restrict__ A, const float* __restrict__ W,
                           const float* __restrict__ bias, float* __restrict__ Out,
                           int N, int K) {
    int lane = threadIdx.x;
    int idx  = lane & 15;
    int half = (lane >> 4) << 1;
    const float* Arow = A + (size_t)(blockIdx.x * 16 + idx) * K;
    const float* Wrow = W + (size_t)(blockIdx.y * 16 + idx) * K;
    v8f acc = {};
    for (int k = 0; k < K; k += 4) {
        v2f a, b;
        a.x = Arow[k + half];     a.y = Arow[k + half + 1];
        b.x = Wrow[k + half];     b.y = Wrow[k + half + 1];
        acc = wmma_f32(a, b, acc);
    }
    int n = blockIdx.y * 16 + idx;
    float bb = bias[n];
#pragma unroll
    for (int r = 0; r < 8; ++r) {
        int m = blockIdx.x * 16 + r + ((lane >> 4) << 3);
        Out[(size_t)m * N + n] = acc[r] + bb;
    }
}

// ----------------------------- K3: qk[b,h,e] = scale * q[b,h,:] @ Wk[h*HD+ :, e]
// grid = (BS/16, EMB/16, NH), block = 32.  (bk cancels in softmax -> dropped)
__global__ void k_qkfold(const float* __restrict__ q, const float* __restrict__ Wk,
                         float* __restrict__ qk) {
    int tm = blockIdx.x, tn = blockIdx.y, h = blockIdx.z;
    int lane = threadIdx.x, idx = lane & 15, half = (lane >> 4) << 1;
    const float* Arow = q + (size_t)(tm * 16 + idx) * EMB + h * HD;
    const float* B    = Wk + (size_t)h * HD * EMB + tn * 16 + idx;
    v8f acc = {};
    for (int k = 0; k < HD; k += 4) {
        v2f a, b;
        a.x = Arow[k + half];               a.y = Arow[k + half + 1];
        b.x = B[(size_t)(k + half) * EMB];  b.y = B[(size_t)(k + half + 1) * EMB];
        acc = wmma_f32(a, b, acc);
    }
    int e = tn * 16 + idx;
#pragma unroll
    for (int r = 0; r < 8; ++r) {
        int bi = tm * 16 + r + ((lane >> 4) << 3);
        qk[((size_t)bi * NH + h) * EMB + e] = acc[r] * QSCALE;
    }
}

// ------------------- K4: scores[b,h,s] = kv_src[b,s,:] . qk[b,h,:]  (kv on the fly)
// grid = (BS, SEQ/16), block = 32. N-dim = 16 heads (exactly one WMMA tile).
__global__ void k_scores(const float* __restrict__ model, const float* __restrict__ know,
                         const int* __restrict__ ids, const int* __restrict__ know_tok,
                         const float* __restrict__ qk, float* __restrict__ scores) {
    int b = blockIdx.x, ts = blockIdx.y;
    int lane = threadIdx.x, idx = lane & 15, half = (lane >> 4) << 1;
    int s = ts * 16 + idx;
    int ktok = know_tok[0];
    const float* Arow = (ids[b * SEQ + s] == ktok)
                      ? (know + (size_t)b * EMB)
                      : (model + ((size_t)b * SEQ + s) * EMB);
    const float* Bh = qk + ((size_t)b * NH + idx) * EMB;   // lane's head column
    v8f acc = {};
    for (int k = 0; k < EMB; k += 4) {
        v2f a, bb;
        a.x  = Arow[k + half]; a.y  = Arow[k + half + 1];
        bb.x = Bh[k + half];   bb.y = Bh[k + half + 1];
        acc = wmma_f32(a, bb, acc);
    }
    int h = idx;
#pragma unroll
    for (int r = 0; r < 8; ++r) {
        int srow = ts * 16 + r + ((lane >> 4) << 3);
        scores[((size_t)b * NH + h) * SEQ + srow] = acc[r];
    }
}

// ---------------------------------------------------------------- K5: softmax
__global__ void k_softmax(float* __restrict__ all) {
    float* p = all + (size_t)blockIdx.x * SEQ;
    __shared__ float red[128];
    int t = threadIdx.x;
    float m = -3.402823466e38f;
    for (int s = t; s < SEQ; s += 128) m = fmaxf(m, p[s]);
    red[t] = m; __syncthreads();
    for (int off = 64; off > 0; off >>= 1) {
        if (t < off) red[t] = fmaxf(red[t], red[t + off]);
        __syncthreads();
    }
    m = red[0];
    __syncthreads();
    float sum = 0.f;
    for (int s = t; s < SEQ; s += 128) { float e = expf(p[s] - m); p[s] = e; sum += e; }
    red[t] = sum; __syncthreads();
    for (int off = 64; off > 0; off >>= 1) {
        if (t < off) red[t] += red[t + off];
        __syncthreads();
    }
    float inv = 1.f / red[0];
    for (int s = t; s < SEQ; s += 128) p[s] *= inv;
}

// ------------------- K6: u[b,h,e] = sum_s attn[b,h,s] * kv_src[b,s,e]
// grid = (BS, EMB/16), block = 32. M-dim = 16 heads (one tile), K = SEQ.
__global__ void k_uacc(const float* __restrict__ attn, const float* __restrict__ model,
                       const float* __restrict__ know, const int* __restrict__ ids,
                       const int* __restrict__ know_tok, float* __restrict__ u) {
    int b = blockIdx.x, te = blockIdx.y;
    int lane = threadIdx.x, idx = lane & 15, half = (lane >> 4) << 1;
    __shared__ unsigned char kf[SEQ];
    int ktok = know_tok[0];
    for (int s = lane; s < SEQ; s += 32)
        kf[s] = (ids[b * SEQ + s] == ktok) ? 1 : 0;
    __syncthreads();
    int e = te * 16 + idx;
    float kval = know[(size_t)b * EMB + e];
    const float* Arow = attn + ((size_t)b * NH + idx) * SEQ;
    const float* Mb   = model + (size_t)b * SEQ * EMB + e;
    v8f acc = {};
    for (int k = 0; k < SEQ; k += 4) {
        int s0 = k + half;
        v2f a, bb;
        a.x  = Arow[s0];  a.y  = Arow[s0 + 1];
        bb.x = kf[s0]     ? kval : Mb[(size_t)s0 * EMB];
        bb.y = kf[s0 + 1] ? kval : Mb[(size_t)(s0 + 1) * EMB];
        acc = wmma_f32(a, bb, acc);
    }
#pragma unroll
    for (int r = 0; r < 8; ++r) {
        int h = r + ((lane >> 4) << 3);
        u[((size_t)b * NH + h) * EMB + e] = acc[r];
    }
}

// ------------------- K7: ctx[b, h*HD+d] = u[b,h,:] @ Wv[h*HD+d,:] + bv[h*HD+d]
// grid = (BS/16, HD/16, NH), block = 32.
__global__ void k_ctx(const float* __restrict__ u, const float* __restrict__ Wv,
                      const float* __restrict__ bv, float* __restrict__ ctx) {
    int tm = blockIdx.x, tn = blockIdx.y, h = blockIdx.z;
    int lane = threadIdx.x, idx = lane & 15, half = (lane >> 4) << 1;
    const float* Arow = u  + ((size_t)(tm * 16 + idx) * NH + h) * EMB;
    const float* Wrow = Wv + (size_t)(h * HD + tn * 16 + idx) * EMB;
    v8f acc = {};
    for (int k = 0; k < EMB; k += 4) {
        v2f a, b;
        a.x = Arow[k + half]; a.y = Arow[k + half + 1];
        b.x = Wrow[k + half]; b.y = Wrow[k + half + 1];
        acc = wmma_f32(a, b, acc);
    }
    int n = h * HD + tn * 16 + idx;
    float bb = bv[n];
#pragma unroll
    for (int r = 0; r < 8; ++r) {
        int m = tm * 16 + r + ((lane >> 4) << 3);
        ctx[(size_t)m * EMB + n] = acc[r] + bb;
    }
}

// ------------------- K8: out = origin_q + sel(strength) * (ctx @ Wo^T + bo)
// grid = (BS/16, EMB/16), block = 32.
__global__ void k_outproj(const float* __restrict__ ctx, const float* __restrict__ Wo,
                          const float* __restrict__ bo, const float* __restrict__ model,
                          const int* __restrict__ mask_pos,
                          const unsigned char* __restrict__ empty,
                          const float* __restrict__ strength, float* __restrict__ out) {
    int lane = threadIdx.x, idx = lane & 15, half = (lane >> 4) << 1;
    const float* Arow = ctx + (size_t)(blockIdx.x * 16 + idx) * EMB;
    const float* Wrow = Wo  + (size_t)(blockIdx.y * 16 + idx) * EMB;
    v8f acc = {};
    for (int k = 0; k < EMB; k += 4) {
        v2f a, b;
        a.x = Arow[k + half]; a.y = Arow[k + half + 1];
        b.x = Wrow[k + half]; b.y = Wrow[k + half + 1];
        acc = wmma_f32(a, b, acc);
    }
    int e = blockIdx.y * 16 + idx;
    float bb = bo[e];
    float st = strength[0];
#pragma unroll
    for (int r = 0; r < 8; ++r) {
        int b = blockIdx.x * 16 + r + ((lane >> 4) << 3);
        float orig = model[((size_t)b * SEQ + mask_pos[b]) * EMB + e];
        float w = empty[b] ? 0.f : st;
        out[(size_t)b * EMB + e] = orig + w * (acc[r] + bb);
    }
}

extern "C" void kernel_launch(void* const* d_in, const int* in_sizes, int n_in,
                              void* d_out, int out_size, void* d_ws, size_t ws_size,
                              hipStream_t stream) {
    (void)in_sizes; (void)n_in; (void)out_size; (void)ws_size;
    const float* model    = (const float*)d_in[0];
    const float* know     = (const float*)d_in[1];
    const float* inW      = (const float*)d_in[2];
    const float* inB      = (const float*)d_in[3];
    const float* outW     = (const float*)d_in[4];
    const float* outB     = (const float*)d_in[5];
    const float* strength = (const float*)d_in[6];
    const int*   ids      = (const int*)d_in[7];
    const unsigned char* empty = (const unsigned char*)d_in[8];
    const int*   know_tok = (const int*)d_in[9];
    const int*   mask_tok = (const int*)d_in[10];
    float* out = (float*)d_out;

    char*  ws       = (char*)d_ws;
    int*   mask_pos = (int*)ws;                              // 64 ints (pad 1KB)
    float* q_in     = (float*)(ws + 1024);                   // BS*E
    float* q        = q_in   + (size_t)BS * EMB;             // BS*E
    float* qk       = q      + (size_t)BS * EMB;             // BS*H*E
    float* scores   = qk     + (size_t)BS * NH * EMB;        // BS*H*S
    float* u        = scores + (size_t)BS * NH * SEQ;        // BS*H*E
    float* ctx      = u      + (size_t)BS * NH * EMB;        // BS*E

    const float* Wq = inW;
    const float* Wk = inW + (size_t)EMB * EMB;
    const float* Wv = inW + 2 * (size_t)EMB * EMB;
    const float* bq = inB;
    const float* bv = inB + 2 * EMB;   // bk cancels in softmax; bv applied in k_ctx

    k_prep   <<<BS, 256, 0, stream>>>(ids, model, know, mask_tok, know_tok, mask_pos, q_in);
    k_gemm_abt<<<dim3(BS/16, EMB/16), 32, 0, stream>>>(q_in, Wq, bq, q, EMB, EMB);
    k_qkfold <<<dim3(BS/16, EMB/16, NH), 32, 0, stream>>>(q, Wk, qk);
    k_scores <<<dim3(BS, SEQ/16), 32, 0, stream>>>(model, know, ids, know_tok, qk, scores);
    k_softmax<<<BS * NH, 128, 0, stream>>>(scores);
    k_uacc   <<<dim3(BS, EMB/16), 32, 0, stream>>>(scores, model, know, ids, know_tok, u);
    k_ctx    <<<dim3(BS/16, HD/16, NH), 32, 0, stream>>>(u, Wv, bv, ctx);
    k_outproj<<<dim3(BS/16, EMB/16), 32, 0, stream>>>(ctx, outW, outB, model, mask_pos,
                                                      empty, strength, out);
}